// RNNBaseline_23003844837555
// MI455X (gfx1250) — compile-verified
//
#include <hip/hip_runtime.h>
#include <cmath>

#define H 3

typedef float v2f __attribute__((ext_vector_type(2)));
typedef float v8f __attribute__((ext_vector_type(8)));

__device__ __forceinline__ float fast_tanh(float v) {
#if __has_builtin(__builtin_amdgcn_tanhf)
    return __builtin_amdgcn_tanhf(v);          // v_tanh_f32 (gfx1250 trans op)
#elif __has_builtin(__builtin_amdgcn_tanh_f32)
    return __builtin_amdgcn_tanh_f32(v);
#else
    return tanhf(v);
#endif
}

// ---------------------------------------------------------------------------
// Phase 1: A[t][n] = sum_k x[t][k] * W_ih[n][k] + b_ih[n] + b_hh[n]
// One wave per 16-timestep tile using V_WMMA_F32_16X16X4_F32:
//   A-operand: 16x4  (M = 16 timesteps, K = input dim padded 3->4)
//   B-operand: 4x16  (B[k][n] = W_ih[n][k], cols n>=3 zero)
//   C-operand: bias broadcast down each used column
// ISA VGPR layouts (32-bit, wave32):
//   A: lanes 0-15 row M=lane, v[0]=K0 v[1]=K1 ; lanes 16-31 row M=lane-16, v[0]=K2 v[1]=K3
//   B: lanes 0-15 col N=lane, v[0]=K0 v[1]=K1 ; lanes 16-31 col N=lane-16, v[0]=K2 v[1]=K3
//   D: lane l -> col N=l&15 ; VGPR r -> row M = r + 8*(l>>4)
// ---------------------------------------------------------------------------
__global__ __launch_bounds__(32)
void rnn_pre_wmma(const float* __restrict__ x,
                  const float* __restrict__ W_ih,
                  const float* __restrict__ b_ih,
                  const float* __restrict__ b_hh,
                  float* __restrict__ A,
                  int T)
{
    const int lane = threadIdx.x;   // 0..31, one wave per block
    const int n    = lane & 15;
    const int hi   = lane >> 4;
    const bool valid = (n < H);
    const int  nc    = valid ? n : 0;

    // B operand (constant across tiles): B[k][n] = W_ih[n][k]
    float wlo = W_ih[nc * H + (hi ? 2 : 0)];  // K = 0 (lanes 0-15) or K = 2 (lanes 16-31)
    float wmid = W_ih[nc * H + 1];            // K = 1
    v2f b;
    b.x = valid ? wlo : 0.0f;
    b.y = (valid && !hi) ? wmid : 0.0f;       // K=3 row is zero padding

    // C operand: combined bias broadcast down each column
    float bias = valid ? (b_ih[nc] + b_hh[nc]) : 0.0f;
    v8f c;
#pragma unroll
    for (int r = 0; r < 8; ++r) c[r] = bias;

    const int ntiles = T / 16;
    for (int tile = blockIdx.x; tile < ntiles; tile += gridDim.x) {
        const float* xr = x + (size_t)(tile * 16 + n) * H;   // row M = n for this lane
        float x0 = xr[hi ? 2 : 0];
        float x1 = xr[1];
        v2f a;
        a.x = x0;
        a.y = hi ? 0.0f : x1;                // K=3 zero padding

        v8f d = __builtin_amdgcn_wmma_f32_16x16x4_f32(
            /*neg_a=*/false, a, /*neg_b=*/false, b,
            /*c_mod=*/(short)0, c, /*reuse_a=*/false, /*reuse_b=*/false);

        if (valid) {
            float* out = A + (size_t)(tile * 16 + hi * 8) * H + n;
#pragma unroll
            for (int r = 0; r < 8; ++r) out[r * H] = d[r];
        }
    }
}

// ---------------------------------------------------------------------------
// Phase 2: inherently sequential scan. Single lane; critical path per step is
// 3 chained v_fmac + 1 v_tanh_f32 (the three hidden components are independent
// ILP chains). a_t loads come from L2-resident workspace, off the chain.
// Readout Linear(3->1) fused here (also off the chain).
// ---------------------------------------------------------------------------
__global__ __launch_bounds__(32)
void rnn_seq(const float* __restrict__ A,
             const float* __restrict__ W_hh,
             const float* __restrict__ W_out,
             const float* __restrict__ b_out,
             float* __restrict__ out,
             int T)
{
    if (threadIdx.x != 0 || blockIdx.x != 0) return;

    const float w00 = W_hh[0], w01 = W_hh[1], w02 = W_hh[2];
    const float w10 = W_hh[3], w11 = W_hh[4], w12 = W_hh[5];
    const float w20 = W_hh[6], w21 = W_hh[7], w22 = W_hh[8];
    const float u0 = W_out[0], u1 = W_out[1], u2 = W_out[2];
    const float bo = b_out[0];

    float h0 = 0.0f, h1 = 0.0f, h2 = 0.0f;
#pragma unroll 4
    for (int t = 0; t < T; ++t) {
        const float a0 = A[(size_t)t * H + 0];
        const float a1 = A[(size_t)t * H + 1];
        const float a2 = A[(size_t)t * H + 2];
        float p0 = fmaf(w02, h2, fmaf(w01, h1, fmaf(w00, h0, a0)));
        float p1 = fmaf(w12, h2, fmaf(w11, h1, fmaf(w10, h0, a1)));
        float p2 = fmaf(w22, h2, fmaf(w21, h1, fmaf(w20, h0, a2)));
        h0 = fast_tanh(p0);
        h1 = fast_tanh(p1);
        h2 = fast_tanh(p2);
        out[t] = fmaf(u2, h2, fmaf(u1, h1, fmaf(u0, h0, bo)));
    }
}

// Fallback: fully fused sequential kernel (no workspace); the input projection
// is computed inline but stays off the h-dependent chain.
__global__ __launch_bounds__(32)
void rnn_seq_fused(const float* __restrict__ x,
                   const float* __restrict__ W_ih,
                   const float* __restrict__ W_hh,
                   const float* __restrict__ b_ih,
                   const float* __restrict__ b_hh,
                   const float* __restrict__ W_out,
                   const float* __restrict__ b_out,
                   float* __restrict__ out,
                   int T)
{
    if (threadIdx.x != 0 || blockIdx.x != 0) return;

    const float v00 = W_ih[0], v01 = W_ih[1], v02 = W_ih[2];
    const float v10 = W_ih[3], v11 = W_ih[4], v12 = W_ih[5];
    const float v20 = W_ih[6], v21 = W_ih[7], v22 = W_ih[8];
    const float w00 = W_hh[0], w01 = W_hh[1], w02 = W_hh[2];
    const float w10 = W_hh[3], w11 = W_hh[4], w12 = W_hh[5];
    const float w20 = W_hh[6], w21 = W_hh[7], w22 = W_hh[8];
    const float bb0 = b_ih[0] + b_hh[0];
    const float bb1 = b_ih[1] + b_hh[1];
    const float bb2 = b_ih[2] + b_hh[2];
    const float u0 = W_out[0], u1 = W_out[1], u2 = W_out[2];
    const float bo = b_out[0];

    float h0 = 0.0f, h1 = 0.0f, h2 = 0.0f;
#pragma unroll 4
    for (int t = 0; t < T; ++t) {
        const float x0 = x[(size_t)t * H + 0];
        const float x1 = x[(size_t)t * H + 1];
        const float x2 = x[(size_t)t * H + 2];
        const float a0 = fmaf(v02, x2, fmaf(v01, x1, fmaf(v00, x0, bb0)));
        const float a1 = fmaf(v12, x2, fmaf(v11, x1, fmaf(v10, x0, bb1)));
        const float a2 = fmaf(v22, x2, fmaf(v21, x1, fmaf(v20, x0, bb2)));
        float p0 = fmaf(w02, h2, fmaf(w01, h1, fmaf(w00, h0, a0)));
        float p1 = fmaf(w12, h2, fmaf(w11, h1, fmaf(w10, h0, a1)));
        float p2 = fmaf(w22, h2, fmaf(w21, h1, fmaf(w20, h0, a2)));
        h0 = fast_tanh(p0);
        h1 = fast_tanh(p1);
        h2 = fast_tanh(p2);
        out[t] = fmaf(u2, h2, fmaf(u1, h1, fmaf(u0, h0, bo)));
    }
}

extern "C" void kernel_launch(void* const* d_in, const int* in_sizes, int n_in,
                              void* d_out, int out_size, void* d_ws, size_t ws_size,
                              hipStream_t stream)
{
    const float* x     = (const float*)d_in[0];
    const float* W_ih  = (const float*)d_in[1];
    const float* W_hh  = (const float*)d_in[2];
    const float* b_ih  = (const float*)d_in[3];
    const float* b_hh  = (const float*)d_in[4];
    const float* W_out = (const float*)d_in[5];
    const float* b_out = (const float*)d_in[6];
    float* out = (float*)d_out;

    const int T = in_sizes[0] / H;
    const size_t need = (size_t)T * H * sizeof(float);

    if ((T % 16) == 0 && ws_size >= need) {
        float* A = (float*)d_ws;
        const int ntiles = T / 16;
        const int blocks = ntiles < 2048 ? ntiles : 2048;
        rnn_pre_wmma<<<blocks, 32, 0, stream>>>(x, W_ih, b_ih, b_hh, A, T);
        rnn_seq<<<1, 32, 0, stream>>>(A, W_hh, W_out, b_out, out, T);
    } else {
        rnn_seq_fused<<<1, 32, 0, stream>>>(x, W_ih, W_hh, b_ih, b_hh,
                                            W_out, b_out, out, T);
    }
}